// GCNEncoder_39659728011299
// MI455X (gfx1250) — compile-verified
//
#include <hip/hip_runtime.h>

#define NN 100000
#define NE 1200000
#define IND 256
#define MIDD 128
#define HID 64
#define NL 5

typedef __attribute__((ext_vector_type(2))) float v2f;
typedef __attribute__((ext_vector_type(8))) float v8f;

// D = A(16x4 f32) * B(4x16 f32) + C(16x16 f32), full fp32 precision (matches reference)
__device__ __forceinline__ v8f wmma4(v2f a, v2f b, v8f c) {
  return __builtin_amdgcn_wmma_f32_16x16x4_f32(false, a, false, b, (short)0, c, false, false);
}

__global__ void k_zero(float* __restrict__ p, int n) {
  int i = blockIdx.x * blockDim.x + threadIdx.x;
  if (i < n) p[i] = 0.0f;
}

__global__ void k_degree(const int* __restrict__ src, const int* __restrict__ dst,
                         float* __restrict__ odeg, float* __restrict__ ideg, int ne) {
  int e = blockIdx.x * blockDim.x + threadIdx.x;
  if (e < ne) {
    atomicAdd(&odeg[src[e]], 1.0f);
    atomicAdd(&ideg[dst[e]], 1.0f);
  }
}

__global__ void k_norm(float* __restrict__ p, int n) {
  int i = blockIdx.x * blockDim.x + threadIdx.x;
  if (i < n) p[i] = rsqrtf(fmaxf(p[i], 1.0f));
}

// Encoder: x = relu(BN(h@W1^T + b1)) @ W2^T + b2 ; writes hcur and initializes acc.
// One wave per 16-row tile, 4 waves per block.
__global__ __launch_bounds__(128) void k_encoder(
    const float* __restrict__ h, const float* __restrict__ w1, const float* __restrict__ b1,
    const float* __restrict__ w2, const float* __restrict__ b2,
    const float* __restrict__ gm, const float* __restrict__ bt,
    const float* __restrict__ mu, const float* __restrict__ vr,
    float* __restrict__ xout, float* __restrict__ accout)
{
  __shared__ float tls[4][16][MIDD + 1];
  const int wave = threadIdx.x >> 5;
  const int lane = threadIdx.x & 31;
  const int m  = lane & 15;   // row (A) or col (B/C) index
  const int kh = lane >> 4;   // k half-select
  const int rowbase = (blockIdx.x * 4 + wave) * 16;  // NN % 16 == 0: tiles are always full

  if (rowbase < NN) {
    const float* hrow = h + (size_t)(rowbase + m) * IND;
    v8f acc1[8] = {};
    for (int k = 0; k < IND; k += 4) {
      v2f a;
      a.x = hrow[k + 2 * kh];
      a.y = hrow[k + 2 * kh + 1];
#pragma unroll
      for (int nt = 0; nt < 8; ++nt) {
        const int n = nt * 16 + m;
        v2f b;                                 // B[k][n] = w1[n][k]  (t = h @ w1^T)
        b.x = w1[n * IND + k + 2 * kh];
        b.y = w1[n * IND + k + 2 * kh + 1];
        acc1[nt] = wmma4(a, b, acc1[nt]);
      }
    }
    // bias + BatchNorm(eval) + ReLU, stage into LDS to re-layout C-tile -> A-operand
#pragma unroll
    for (int nt = 0; nt < 8; ++nt) {
      const int col = nt * 16 + m;
      const float K  = gm[col] * rsqrtf(vr[col] + 1e-5f);
      const float sh = (b1[col] - mu[col]) * K + bt[col];
#pragma unroll
      for (int r = 0; r < 8; ++r) {
        const int row = r + 8 * kh;            // C layout: row = r + 8*(lane/16)
        tls[wave][row][col] = fmaxf(acc1[nt][r] * K + sh, 0.0f);
      }
    }
  }
  __syncthreads();
  if (rowbase >= NN) return;

  v8f acc2[4] = {};
  for (int k = 0; k < MIDD; k += 4) {
    v2f a;
    a.x = tls[wave][m][k + 2 * kh];
    a.y = tls[wave][m][k + 2 * kh + 1];
#pragma unroll
    for (int nt = 0; nt < 4; ++nt) {
      const int n = nt * 16 + m;
      v2f b;                                   // B[k][n] = w2[n][k]
      b.x = w2[n * MIDD + k + 2 * kh];
      b.y = w2[n * MIDD + k + 2 * kh + 1];
      acc2[nt] = wmma4(a, b, acc2[nt]);
    }
  }
#pragma unroll
  for (int nt = 0; nt < 4; ++nt) {
    const int col = nt * 16 + m;
    const float bias = b2[col];
#pragma unroll
    for (int r = 0; r < 8; ++r) {
      const size_t row = (size_t)(rowbase + r + 8 * kh);
      const float v = acc2[nt][r] + bias;
      xout[row * HID + col]   = v;
      accout[row * HID + col] = v;   // acc starts at x
    }
  }
}

// hw = (hcur * norm_out[:,None]) @ W   with W in DGL [in,out] layout (B = W directly)
__global__ __launch_bounds__(256) void k_gcn_gemm(
    const float* __restrict__ hc, const float* __restrict__ W,
    const float* __restrict__ nrm, float* __restrict__ hw)
{
  const int wave = threadIdx.x >> 5;
  const int lane = threadIdx.x & 31;
  const int m  = lane & 15;
  const int kh = lane >> 4;
  const int rowbase = (blockIdx.x * 8 + wave) * 16;
  if (rowbase >= NN) return;

  const int row = rowbase + m;
  const float s = nrm[row];
  const float* hrow = hc + (size_t)row * HID;
  v8f acc[4] = {};
  for (int k = 0; k < HID; k += 4) {
    v2f a;
    a.x = hrow[k + 2 * kh] * s;
    a.y = hrow[k + 2 * kh + 1] * s;
#pragma unroll
    for (int nt = 0; nt < 4; ++nt) {
      const int n = nt * 16 + m;
      v2f b;                                   // B[k][n] = W[k*HID + n]
      b.x = W[(k + 2 * kh) * HID + n];
      b.y = W[(k + 2 * kh + 1) * HID + n];
      acc[nt] = wmma4(a, b, acc[nt]);
    }
  }
#pragma unroll
  for (int nt = 0; nt < 4; ++nt) {
    const int col = nt * 16 + m;
#pragma unroll
    for (int r = 0; r < 8; ++r) {
      hw[(size_t)(rowbase + r + 8 * kh) * HID + col] = acc[nt][r];
    }
  }
}

// agg[dst] += hw[src]; 64 consecutive threads cover one edge row (coalesced gather+scatter)
__global__ void k_scatter(const float* __restrict__ hw, const int* __restrict__ src,
                          const int* __restrict__ dst, float* __restrict__ agg) {
  unsigned i = blockIdx.x * blockDim.x + threadIdx.x;
  if (i < (unsigned)NE * HID) {
    const int e = i >> 6;
    const int c = i & 63;
    atomicAdd(&agg[(size_t)dst[e] * HID + c], hw[(size_t)src[e] * HID + c]);
  }
}

__global__ void k_post(const float* __restrict__ agg, const float* __restrict__ nin,
                       float* __restrict__ hc, float* __restrict__ acc, int relu) {
  int i = blockIdx.x * blockDim.x + threadIdx.x;
  if (i < NN * HID) {
    float v = agg[i] * nin[i >> 6];
    if (relu) v = fmaxf(v, 0.0f);
    hc[i] = v;
    acc[i] += v;
  }
}

__global__ void k_scale(float* __restrict__ p, int n, float s) {
  int i = blockIdx.x * blockDim.x + threadIdx.x;
  if (i < n) p[i] *= s;
}

extern "C" void kernel_launch(void* const* d_in, const int* in_sizes, int n_in,
                              void* d_out, int out_size, void* d_ws, size_t ws_size,
                              hipStream_t stream) {
  (void)in_sizes; (void)n_in; (void)out_size; (void)ws_size;
  const float* h   = (const float*)d_in[0];
  const int*   src = (const int*)d_in[1];
  const int*   dst = (const int*)d_in[2];
  const float* w1  = (const float*)d_in[3];
  const float* b1  = (const float*)d_in[4];
  const float* w2  = (const float*)d_in[5];
  const float* b2  = (const float*)d_in[6];
  const float* gm  = (const float*)d_in[7];
  const float* bt  = (const float*)d_in[8];
  const float* mu  = (const float*)d_in[9];
  const float* vr  = (const float*)d_in[10];
  const float* Wg  = (const float*)d_in[11];
  float* out = (float*)d_out;

  float* ws       = (float*)d_ws;
  float* norm_out = ws;                        // NN floats
  float* norm_in  = ws + NN;                   // NN floats (contiguous after norm_out)
  float* hc       = ws + 2 * NN;               // NN*HID
  float* hw       = hc + (size_t)NN * HID;     // NN*HID
  float* agg      = hw + (size_t)NN * HID;     // NN*HID

  // degree -> D^{-1/2} norms (zeroed every call: deterministic under graph replay)
  k_zero<<<(2 * NN + 255) / 256, 256, 0, stream>>>(norm_out, 2 * NN);
  k_degree<<<(NE + 255) / 256, 256, 0, stream>>>(src, dst, norm_out, norm_in, NE);
  k_norm<<<(2 * NN + 255) / 256, 256, 0, stream>>>(norm_out, 2 * NN);

  // encoder: writes hc and seeds the accumulator in d_out
  k_encoder<<<(NN + 63) / 64, 128, 0, stream>>>(h, w1, b1, w2, b2, gm, bt, mu, vr, hc, out);

  for (int l = 0; l < NL; ++l) {
    k_zero<<<(NN * HID + 255) / 256, 256, 0, stream>>>(agg, NN * HID);
    k_gcn_gemm<<<(NN + 127) / 128, 256, 0, stream>>>(hc, Wg + (size_t)l * HID * HID,
                                                     norm_out, hw);
    k_scatter<<<((unsigned)NE * HID + 255) / 256, 256, 0, stream>>>(hw, src, dst, agg);
    k_post<<<(NN * HID + 255) / 256, 256, 0, stream>>>(agg, norm_in, hc, out,
                                                       (l != NL - 1) ? 1 : 0);
  }
  k_scale<<<(NN * HID + 255) / 256, 256, 0, stream>>>(out, NN * HID, 1.0f / (NL + 1));
}